// T_Mamba_5446018531503
// MI455X (gfx1250) — compile-verified
//
#include <hip/hip_runtime.h>
#include <math.h>

// ---------------------------------------------------------------------------
// Problem constants (from reference): B=4, Cin=64, Chid=96, H=W=64, N=16, R=6,
// K=4 directions, Dn=192, L=4096.
// ---------------------------------------------------------------------------
static constexpr int Bsz  = 4;
static constexpr int Cin  = 64;
static constexpr int Chid = 96;
static constexpr int Hh   = 64;
static constexpr int Ww   = 64;
static constexpr int Lsp  = Hh * Ww;          // 4096
static constexpr int Rows = Bsz * Lsp;        // 16384 (= B*H*W)
static constexpr int Nst  = 16;               // state size N
static constexpr int Rrank= 6;                // R
static constexpr int Kdir = 4;                // scan directions
static constexpr int Dn   = 192;              // 2*Chid
static constexpr int Cproj= Rrank + 2 * Nst;  // 38

typedef __attribute__((ext_vector_type(2))) float v2f;
typedef __attribute__((ext_vector_type(8))) float v8f;

#define DEV __device__ __forceinline__

DEV float siluf(float x)     { return x / (1.f + __expf(-x)); }
DEV float sigmf(float x)     { return 1.f / (1.f + __expf(-x)); }
DEV float softplusf(float x) { return (x > 20.f) ? x : log1pf(__expf(x)); }

// ---------------------------------------------------------------------------
// Wave-level 16x16 fp32 WMMA tile: C(16x16) += A(16xK) * B(Kx16).
// Operand layout per CDNA5 ISA (16x16x4 f32):
//   A: lane holds row M = lane%16, elements K = k + 2*(lane/16) + {0,1}
//   B: lane holds col N = lane%16, elements K = k + 2*(lane/16) + {0,1}
//   C/D: vgpr v -> row M = v + 8*(lane/16), col N = lane%16
// ---------------------------------------------------------------------------
template <typename FA, typename FB>
DEV v8f wmma_tile_f32(int K, FA la, FB lb) {
  const int lane = threadIdx.x & 31;
  const int half = lane >> 4;
  v8f acc = {};
  for (int k = 0; k < K; k += 4) {
    const int kk = k + 2 * half;
    v2f a, b;
    a[0] = la(lane & 15, kk);
    a[1] = la(lane & 15, kk + 1);
    b[0] = lb(kk,     lane & 15);
    b[1] = lb(kk + 1, lane & 15);
    acc = __builtin_amdgcn_wmma_f32_16x16x4_f32(false, a, false, b,
                                                (short)0, acc, false, false);
  }
  return acc;
}

// ---------------------------------------------------------------------------
// 0) NCHW -> (row, c) transpose of pre/post + |post-pre|
// ---------------------------------------------------------------------------
__global__ void prep_kernel(const float* __restrict__ pre,
                            const float* __restrict__ post,
                            float* __restrict__ diffT,
                            float* __restrict__ preT,
                            float* __restrict__ postT) {
  int idx = blockIdx.x * blockDim.x + threadIdx.x;   // over Rows*Cin
  if (idx >= Rows * Cin) return;
  int c = idx % Cin;
  int row = idx / Cin;                                // b*L + l
  int b = row / Lsp, l = row % Lsp;
  size_t src = ((size_t)b * Cin + c) * Lsp + l;
  float p0 = pre[src], p1 = post[src];
  preT[idx] = p0;
  postT[idx] = p1;
  diffT[idx] = fabsf(p1 - p0);
}

// ---------------------------------------------------------------------------
// 1) Generic GEMM: C[M,N] = A[M,K] * W[N,K]^T, fused epilogue.
//    ep=0: +bias   ep=1: +bias,BN,SiLU   ep=2: +bias,BN,SiLU,sigmoid
//    Requires M%16==0, N%16==0, K%4==0 (true for all uses).
// ---------------------------------------------------------------------------
__global__ void gemm_kernel(const float* __restrict__ A,
                            const float* __restrict__ W,
                            const float* __restrict__ bias,
                            const float* __restrict__ bng,
                            const float* __restrict__ bnb,
                            float* __restrict__ C,
                            int M, int N, int K, int ep) {
  int wav = blockIdx.x * (blockDim.x >> 5) + (threadIdx.x >> 5);
  int tn = N >> 4;
  int tiles = (M >> 4) * tn;
  if (wav >= tiles) return;                   // wave-uniform: EXEC stays full
  int m0 = (wav / tn) << 4;
  int n0 = (wav % tn) << 4;

  auto la = [&](int r, int k) { return A[(size_t)(m0 + r) * K + k]; };
  auto lb = [&](int k, int c) { return W[(size_t)(n0 + c) * K + k]; };
  v8f acc = wmma_tile_f32(K, la, lb);

  int lane = threadIdx.x & 31, half = lane >> 4, lm = lane & 15;
  int n = n0 + lm;
  float bs = bias ? bias[n] : 0.f;
  float gi = 0.f, bb = 0.f;
  if (ep >= 1) { gi = bng[n] * rsqrtf(1.f + 1e-5f); bb = bnb[n]; }
#pragma unroll
  for (int v = 0; v < 8; ++v) {
    int m = m0 + v + 8 * half;
    float x = acc[v] + bs;
    if (ep >= 1) { x = x * gi + bb; x = siluf(x); }
    if (ep == 2) { x = sigmf(x); }
    C[(size_t)m * N + n] = x;
  }
}

// ---------------------------------------------------------------------------
// 2) Row LayerNorm (wave per row), optional *silu(Z[row, zoff+c]) gate.
// ---------------------------------------------------------------------------
__global__ void ln_kernel(float* __restrict__ X,
                          const float* __restrict__ g,
                          const float* __restrict__ bta,
                          const float* __restrict__ Z, int zld, int zoff,
                          int C) {
  int wav = blockIdx.x * (blockDim.x >> 5) + (threadIdx.x >> 5);
  if (wav >= Rows) return;
  int lane = threadIdx.x & 31;
  float s = 0.f, ss = 0.f;
  for (int c = lane; c < C; c += 32) {
    float x = X[(size_t)wav * C + c];
    s += x; ss += x * x;
  }
  for (int m = 16; m >= 1; m >>= 1) {
    s  += __shfl_xor(s,  m, 32);
    ss += __shfl_xor(ss, m, 32);
  }
  float mean = s / C;
  float var  = ss / C - mean * mean;
  float inv  = rsqrtf(var + 1e-5f);
  for (int c = lane; c < C; c += 32) {
    float x = (X[(size_t)wav * C + c] - mean) * inv * g[c] + bta[c];
    if (Z) x *= siluf(Z[(size_t)wav * zld + zoff + c]);
    X[(size_t)wav * C + c] = x;
  }
}

// ---------------------------------------------------------------------------
// 3) Depthwise 3x3 conv + bias + SiLU.  In: xz[row, d] (ld=2*Dn, first half).
//    Out: xc[b, d, l]  (channel-major for the scan / x_proj consumers).
// ---------------------------------------------------------------------------
__global__ void dwconv_kernel(const float* __restrict__ xz,
                              const float* __restrict__ cw,
                              const float* __restrict__ cb,
                              float* __restrict__ xc) {
  int idx = blockIdx.x * blockDim.x + threadIdx.x;  // over Bsz*Dn*Lsp
  if (idx >= Bsz * Dn * Lsp) return;
  int l = idx % Lsp;
  int d = (idx / Lsp) % Dn;
  int b = idx / (Lsp * Dn);
  int h = l / Ww, w = l % Ww;
  float acc = cb[d];
#pragma unroll
  for (int i = 0; i < 3; ++i) {
    int h2 = h + i - 1;
    if (h2 < 0 || h2 >= Hh) continue;
#pragma unroll
    for (int j = 0; j < 3; ++j) {
      int w2 = w + j - 1;
      if (w2 < 0 || w2 >= Ww) continue;
      acc += cw[d * 9 + i * 3 + j] *
             xz[((size_t)b * Lsp + h2 * Ww + w2) * (2 * Dn) + d];
    }
  }
  xc[((size_t)b * Dn + d) * Lsp + l] = siluf(acc);
}

// Direction-k view of xs: xs[b,k,d,l] addressed straight out of xc.
DEV float xs_load(const float* __restrict__ xc, int b, int k, int d, int l) {
  int lx;
  if (k == 0)      lx = l;
  else if (k == 1) lx = (l % Hh) * Ww + (l / Hh);           // transpose view
  else if (k == 2) lx = Lsp - 1 - l;                        // flip
  else { int lf = Lsp - 1 - l; lx = (lf % Hh) * Ww + (lf / Hh); }
  return xc[((size_t)b * Dn + d) * Lsp + lx];
}

// ---------------------------------------------------------------------------
// 4) x_proj: for each (b,k): x_dbl[c,l] = sum_d x_proj_w[k,c,d] * xs[b,k,d,l].
//    M=38 (padded to 48 tiles of 16), N=L, K=Dn.  WMMA f32.
// ---------------------------------------------------------------------------
__global__ void xproj_kernel(const float* __restrict__ xc,
                             const float* __restrict__ xw,
                             float* __restrict__ xdbl) {
  const int tnL = Lsp >> 4;                 // 256
  const int tilesPerBK = 3 * tnL;           // ceil(38/16)=3 row tiles
  int wav = blockIdx.x * (blockDim.x >> 5) + (threadIdx.x >> 5);
  if (wav >= Bsz * Kdir * tilesPerBK) return;
  int bk = wav / tilesPerBK, t = wav % tilesPerBK;
  int k = bk % Kdir, b = bk / Kdir;
  int m0 = (t / tnL) << 4;
  int n0 = (t % tnL) << 4;
  const float* Wk = xw + (size_t)k * Cproj * Dn;

  auto la = [&](int r, int kk) {
    int m = m0 + r;
    return (m < Cproj) ? Wk[(size_t)m * Dn + kk] : 0.f;
  };
  auto lb = [&](int kk, int c) { return xs_load(xc, b, k, kk, n0 + c); };
  v8f acc = wmma_tile_f32(Dn, la, lb);

  int lane = threadIdx.x & 31, half = lane >> 4, lm = lane & 15;
#pragma unroll
  for (int v = 0; v < 8; ++v) {
    int m = m0 + v + 8 * half;
    if (m < Cproj)
      xdbl[(((size_t)(b * Kdir + k)) * Cproj + m) * Lsp + n0 + lm] = acc[v];
  }
}

// ---------------------------------------------------------------------------
// 5) Selective scan. Block = (b, k, 16 d-channels); thread = (d_local, n).
//    Rank-6 dt projection folded in; 16-lane shuffle reduce for h·C.
// ---------------------------------------------------------------------------
__global__ void scan_kernel(const float* __restrict__ xc,
                            const float* __restrict__ xdbl,
                            const float* __restrict__ dtw,
                            const float* __restrict__ dtbp,
                            const float* __restrict__ Alog,
                            const float* __restrict__ DsP,
                            float* __restrict__ ys) {
  const int dblocks = Dn / 16;             // 12
  int blk = blockIdx.x;                    // Bsz*Kdir*12 = 192
  int dblk = blk % dblocks;
  int k = (blk / dblocks) % Kdir;
  int b = blk / (dblocks * Kdir);
  int t = threadIdx.x;                     // 256
  int dl = t >> 4, n = t & 15;
  int d = dblk * 16 + dl;

  float Ad  = -__expf(Alog[((size_t)(k * Dn) + d) * Nst + n]);
  float dtb = dtbp[k * Dn + d];
  float Dv  = DsP[k * Dn + d];
  float w0 = dtw[((size_t)(k * Dn) + d) * Rrank + 0];
  float w1 = dtw[((size_t)(k * Dn) + d) * Rrank + 1];
  float w2 = dtw[((size_t)(k * Dn) + d) * Rrank + 2];
  float w3 = dtw[((size_t)(k * Dn) + d) * Rrank + 3];
  float w4 = dtw[((size_t)(k * Dn) + d) * Rrank + 4];
  float w5 = dtw[((size_t)(k * Dn) + d) * Rrank + 5];

  const float* xd  = xdbl + ((size_t)(b * Kdir + k)) * Cproj * Lsp;
  const float* xB  = xd + (size_t)(Rrank + n) * Lsp;
  const float* xC  = xd + (size_t)(Rrank + Nst + n) * Lsp;
  float* yrow = ys + (((size_t)(b * Kdir + k)) * Dn + d) * Lsp;

  float h = 0.f;
  for (int l = 0; l < Lsp; ++l) {
    float x_t = xs_load(xc, b, k, d, l);
    float dv = dtb + w0 * xd[0 * Lsp + l] + w1 * xd[1 * Lsp + l]
                   + w2 * xd[2 * Lsp + l] + w3 * xd[3 * Lsp + l]
                   + w4 * xd[4 * Lsp + l] + w5 * xd[5 * Lsp + l];
    float delta = softplusf(dv);
    float Bt = xB[l], Ct = xC[l];
    h = __expf(delta * Ad) * h + (delta * x_t) * Bt;
    float p = h * Ct;
    p += __shfl_xor(p, 8, 32);
    p += __shfl_xor(p, 4, 32);
    p += __shfl_xor(p, 2, 32);
    p += __shfl_xor(p, 1, 32);
    if (n == 0) yrow[l] = p + Dv * x_t;
  }
}

// ---------------------------------------------------------------------------
// 6) Combine 4 directions (flip / transpose-back) -> y[row, d].
// ---------------------------------------------------------------------------
__global__ void combine_kernel(const float* __restrict__ ys,
                               float* __restrict__ y) {
  int idx = blockIdx.x * blockDim.x + threadIdx.x;  // Rows*Dn
  if (idx >= Rows * Dn) return;
  int d = idx % Dn;
  int row = idx / Dn;
  int b = row / Lsp, t = row % Lsp;
  int tT = (t % Ww) * Hh + (t / Ww);
  const float* base = ys + (size_t)b * Kdir * Dn * Lsp;
  float v0 = base[((size_t)0 * Dn + d) * Lsp + t];
  float v1 = base[((size_t)1 * Dn + d) * Lsp + tT];
  float v2 = base[((size_t)2 * Dn + d) * Lsp + (Lsp - 1 - t)];
  float v3 = base[((size_t)3 * Dn + d) * Lsp + (Lsp - 1 - tT)];
  y[idx] = v0 + v1 + v2 + v3;
}

// ---------------------------------------------------------------------------
// 7) Final: out(NCHW) = (up + diff) * g1 * g2
// ---------------------------------------------------------------------------
__global__ void final_kernel(const float* __restrict__ up,
                             const float* __restrict__ diffT,
                             const float* __restrict__ g1,
                             const float* __restrict__ g2,
                             float* __restrict__ out) {
  int idx = blockIdx.x * blockDim.x + threadIdx.x;  // Rows*Cin
  if (idx >= Rows * Cin) return;
  int c = idx % Cin;
  int row = idx / Cin;
  int b = row / Lsp, l = row % Lsp;
  float v = (up[idx] + diffT[idx]) * g1[idx] * g2[idx];
  out[((size_t)b * Cin + c) * Lsp + l] = v;
}

// ---------------------------------------------------------------------------
// Host-side launch
// ---------------------------------------------------------------------------
extern "C" void kernel_launch(void* const* d_in, const int* in_sizes, int n_in,
                              void* d_out, int out_size, void* d_ws,
                              size_t ws_size, hipStream_t stream) {
  (void)in_sizes; (void)n_in; (void)out_size; (void)ws_size;

  const float* pre        = (const float*)d_in[0];
  const float* post       = (const float*)d_in[1];
  const float* prepost_w  = (const float*)d_in[2];
  const float* prepost_b  = (const float*)d_in[3];
  const float* prepost_g  = (const float*)d_in[4];
  const float* prepost_bb = (const float*)d_in[5];
  const float* down_w     = (const float*)d_in[6];
  const float* down_b     = (const float*)d_in[7];
  const float* down_g     = (const float*)d_in[8];
  const float* down_bb    = (const float*)d_in[9];
  const float* up_w       = (const float*)d_in[10];
  const float* up_b       = (const float*)d_in[11];
  const float* up_g       = (const float*)d_in[12];
  const float* up_bb      = (const float*)d_in[13];
  const float* pe_w       = (const float*)d_in[14];
  const float* pe_b       = (const float*)d_in[15];
  const float* pe_ln_g    = (const float*)d_in[16];
  const float* pe_ln_b    = (const float*)d_in[17];
  const float* in_proj_w  = (const float*)d_in[18];
  const float* in_proj_b  = (const float*)d_in[19];
  const float* conv_w     = (const float*)d_in[20];
  const float* conv_b     = (const float*)d_in[21];
  const float* x_proj_w   = (const float*)d_in[22];
  const float* dt_w       = (const float*)d_in[23];
  const float* dt_b       = (const float*)d_in[24];
  const float* A_log      = (const float*)d_in[25];
  const float* Ds         = (const float*)d_in[26];
  const float* out_ln_g   = (const float*)d_in[27];
  const float* out_ln_b   = (const float*)d_in[28];
  const float* out_proj_w = (const float*)d_in[29];
  const float* out_proj_b = (const float*)d_in[30];
  float* out = (float*)d_out;

  float* ws = (float*)d_ws;
  size_t off = 0;
  float* diffT = ws + off; off += (size_t)Rows * Cin;
  float* preT  = ws + off; off += (size_t)Rows * Cin;
  float* postT = ws + off; off += (size_t)Rows * Cin;
  float* d1    = ws + off; off += (size_t)Rows * Chid;
  float* d2    = ws + off; off += (size_t)Rows * Chid;
  float* xz    = ws + off; off += (size_t)Rows * 2 * Dn;
  float* xc    = ws + off; off += (size_t)Bsz * Dn * Lsp;
  float* xdbl  = ws + off; off += (size_t)Bsz * Kdir * Cproj * Lsp;
  float* ysb   = ws + off; off += (size_t)Bsz * Kdir * Dn * Lsp;
  float* ycmb  = ws + off; off += (size_t)Rows * Dn;
  float* d3    = ws + off; off += (size_t)Rows * Chid;
  float* upb   = ws + off; off += (size_t)Rows * Cin;
  float* g1    = ws + off; off += (size_t)Rows * Cin;
  float* g2    = ws + off; off += (size_t)Rows * Cin;

  const int TB = 256;
  auto gemmBlocks = [](int M, int N) { return ((M >> 4) * (N >> 4) + 7) / 8; };

  // 0) transpose + diff
  prep_kernel<<<(Rows * Cin + TB - 1) / TB, TB, 0, stream>>>(pre, post, diffT,
                                                             preT, postT);
  // 1) down: silu(bn(diff @ down_w^T + b))
  gemm_kernel<<<gemmBlocks(Rows, Chid), TB, 0, stream>>>(
      diffT, down_w, down_b, down_g, down_bb, d1, Rows, Chid, Cin, 1);
  // 2) pe: d1 @ pe_w^T + b
  gemm_kernel<<<gemmBlocks(Rows, Chid), TB, 0, stream>>>(
      d1, pe_w, pe_b, nullptr, nullptr, d2, Rows, Chid, Chid, 0);
  // 3) pe LN
  ln_kernel<<<(Rows + 7) / 8, TB, 0, stream>>>(d2, pe_ln_g, pe_ln_b, nullptr,
                                               0, 0, Chid);
  // 4) in_proj -> xz (xc | z)
  gemm_kernel<<<gemmBlocks(Rows, 2 * Dn), TB, 0, stream>>>(
      d2, in_proj_w, in_proj_b, nullptr, nullptr, xz, Rows, 2 * Dn, Chid, 0);
  // 5) depthwise conv + silu -> xc (b,d,l)
  dwconv_kernel<<<(Bsz * Dn * Lsp + TB - 1) / TB, TB, 0, stream>>>(xz, conv_w,
                                                                   conv_b, xc);
  // 6) x_proj -> x_dbl (b,k,38,l)
  {
    int waves = Bsz * Kdir * 3 * (Lsp >> 4);
    xproj_kernel<<<(waves + 7) / 8, TB, 0, stream>>>(xc, x_proj_w, xdbl);
  }
  // 7) selective scan -> ys (b,k,d,l)
  scan_kernel<<<Bsz * Kdir * (Dn / 16), TB, 0, stream>>>(
      xc, xdbl, dt_w, dt_b, A_log, Ds, ysb);
  // 8) combine 4 directions -> y (row, d)
  combine_kernel<<<(Rows * Dn + TB - 1) / TB, TB, 0, stream>>>(ysb, ycmb);
  // 9) out LN + silu(z) gate (in place)
  ln_kernel<<<(Rows + 7) / 8, TB, 0, stream>>>(ycmb, out_ln_g, out_ln_b, xz,
                                               2 * Dn, Dn, Dn);
  // 10) out_proj -> d3
  gemm_kernel<<<gemmBlocks(Rows, Chid), TB, 0, stream>>>(
      ycmb, out_proj_w, out_proj_b, nullptr, nullptr, d3, Rows, Chid, Dn, 0);
  // 11) up: silu(bn(d3 @ up_w^T + b))
  gemm_kernel<<<gemmBlocks(Rows, Cin), TB, 0, stream>>>(
      d3, up_w, up_b, up_g, up_bb, upb, Rows, Cin, Chid, 1);
  // 12) gates
  gemm_kernel<<<gemmBlocks(Rows, Cin), TB, 0, stream>>>(
      preT, prepost_w, prepost_b, prepost_g, prepost_bb, g1, Rows, Cin, Cin, 2);
  gemm_kernel<<<gemmBlocks(Rows, Cin), TB, 0, stream>>>(
      postT, prepost_w, prepost_b, prepost_g, prepost_bb, g2, Rows, Cin, Cin, 2);
  // 13) final combine -> NCHW output
  final_kernel<<<(Rows * Cin + TB - 1) / TB, TB, 0, stream>>>(upb, diffT, g1,
                                                              g2, out);
}